// LocalEnergyOpt_90168543412913
// MI455X (gfx1250) — compile-verified
//
#include <hip/hip_runtime.h>

#define B_C     64
#define NA_C    4096
#define NB_C    4096
#define NANG_C  8192
#define NT_C    12288
#define ROWS_C  (5 * NT_C)   // 61440
#define SPLIT   16
#define THREADS 512
#define WAVES   (THREADS / 32)
#define TCHUNK  (NT_C / SPLIT)          // 768 torsions per block
#define TROWS   (TCHUNK * 5)            // 3840 feature rows of column 8
#define TROWS_W (TROWS / WAVES)         // 240 rows per wave
#define CROWS   (3 * NA_C)              // 12288 coord rows
#define CROWS_W (CROWS / WAVES)         // 768 rows per wave

#if defined(__has_builtin)
#if __has_builtin(__builtin_amdgcn_tensor_load_to_lds)
#define HAVE_TDM 1
#endif
#endif
#ifndef HAVE_TDM
#define HAVE_TDM 0
#endif

typedef unsigned int u32x4 __attribute__((ext_vector_type(4)));
typedef int          i32x4 __attribute__((ext_vector_type(4)));
typedef int          i32x8 __attribute__((ext_vector_type(8)));

struct F3 { float x, y, z; };
__device__ __forceinline__ F3 f3sub(F3 a, F3 b) { return {a.x - b.x, a.y - b.y, a.z - b.z}; }
__device__ __forceinline__ F3 f3cross(F3 a, F3 b) {
    return {a.y * b.z - a.z * b.y, a.z * b.x - a.x * b.z, a.x * b.y - a.y * b.x};
}
__device__ __forceinline__ float f3dot(F3 a, F3 b) { return a.x * b.x + a.y * b.y + a.z * b.z; }
__device__ __forceinline__ F3 f3scale(F3 a, float s) { return {a.x * s, a.y * s, a.z * s}; }

#if HAVE_TDM
// DMA `rows` elements of a stride-9-float column starting at global byte
// address `ga` into contiguous LDS at byte offset `lds_off` (compacting).
__device__ __forceinline__ void tdm_column_to_lds(unsigned lds_off,
                                                  unsigned long long ga,
                                                  int rows)
{
    // D# group 0: count=1 | lds_addr | global_addr[56:0] | type=2
    u32x4 g0;
    g0[0] = 1u;
    g0[1] = lds_off;
    g0[2] = (unsigned)(ga & 0xFFFFFFFFull);
    g0[3] = ((unsigned)(ga >> 32) & 0x01FFFFFFu) | 0x80000000u;
    // D# group 1: data_size=4B; tensor_dim0=9, tensor_dim1=rows;
    // tile 1 x rows; tensor_dim0_stride=9
    i32x8 g1;
    g1[0] = (int)(2u << 16);               // workgroup_mask=0 | data_size=2 (4B)
    g1[1] = (int)(9u << 16);               // tensor_dim0 lo16 = 9
    g1[2] = (int)((unsigned)rows << 16);   // tensor_dim0 hi=0 | tensor_dim1 lo16
    g1[3] = (int)(1u << 16);               // tensor_dim1 hi=0 | tile_dim0=1
    g1[4] = rows;                          // tile_dim1 | tile_dim2=0
    g1[5] = 9;                             // tensor_dim0_stride lo32
    g1[6] = 0;
    g1[7] = 0;
    i32x4 g2 = {0, 0, 0, 0};
    i32x4 g3 = {0, 0, 0, 0};
#if defined(__clang_major__) && (__clang_major__ >= 23)
    i32x8 g4 = {0, 0, 0, 0, 0, 0, 0, 0};
    __builtin_amdgcn_tensor_load_to_lds(g0, g1, g2, g3, g4, 0);
#else
    __builtin_amdgcn_tensor_load_to_lds(g0, g1, g2, g3, 0);
#endif
}
#endif

__global__ void __launch_bounds__(THREADS)
energy_kernel(const float* __restrict__ features,
              const int*   __restrict__ lengths,
              const float* __restrict__ bond_type,
              const float* __restrict__ angle_type,
              const float* __restrict__ tor_type,
              float*       __restrict__ ws)
{
    __shared__ float sc[CROWS];          // 48 KB: compacted coords for this batch
    __shared__ float st[TROWS];          // 15 KB: compacted torsion rows for this chunk
    __shared__ float red[WAVES][3];

    const int tid = (int)threadIdx.x;
    const int b   = (int)blockIdx.x / SPLIT;
    const int s   = (int)blockIdx.x % SPLIT;
    const float* feat = features + (size_t)b * ROWS_C * 9;
    const int ti0 = s * TCHUNK;          // first torsion handled by this block

#if HAVE_TDM
    {
        const int wv = __builtin_amdgcn_readfirstlane(tid >> 5);
        const unsigned long long fbase = (unsigned long long)(size_t)feat;
        // coords: column 5, rows [wv*768, wv*768+768)
        tdm_column_to_lds((unsigned)(unsigned long long)(void*)&sc[0] + (unsigned)(wv * CROWS_W * 4),
                          fbase + (unsigned long long)(wv * CROWS_W) * 36ull + 20ull,
                          CROWS_W);
        // torsion rows: column 8, rows [5*ti0 + wv*240, +240)
        tdm_column_to_lds((unsigned)(unsigned long long)(void*)&st[0] + (unsigned)(wv * TROWS_W * 4),
                          fbase + (unsigned long long)(5 * ti0 + wv * TROWS_W) * 36ull + 32ull,
                          TROWS_W);
        __builtin_amdgcn_s_wait_tensorcnt(0);
    }
#else
    for (int r = tid; r < CROWS; r += THREADS) sc[r] = feat[(size_t)r * 9 + 5];
    for (int r = tid; r < TROWS; r += THREADS) st[r] = feat[(size_t)(5 * ti0 + r) * 9 + 8];
#endif
    __syncthreads();

    const int nb   = lengths[b * 9 + 6] / 3;
    const int nang = lengths[b * 9 + 7] / 4;
    const int nt   = lengths[b * 9 + 8] / 5;

    float eb = 0.0f, ea = 0.0f, et = 0.0f;

    // ---- Bonds: k * (r - r0)^2 ---- (indices streamed from global)
    {
        const int chunk = NB_C / SPLIT;   // 256
        const int i0 = s * chunk;
        for (int i = i0 + tid; i < i0 + chunk; i += THREADS) {
            const float* rp = feat + (size_t)(3 * i) * 9 + 6;
            int ai = (int)rp[0];
            int aj = (int)rp[9];
            int t  = (int)rp[18];
            if (i < nb) {
                F3 pi = {sc[3 * ai], sc[3 * ai + 1], sc[3 * ai + 2]};
                F3 pj = {sc[3 * aj], sc[3 * aj + 1], sc[3 * aj + 2]};
                F3 d  = f3sub(pi, pj);
                float r  = sqrtf(f3dot(d, d));
                float dr = r - bond_type[2 * t];
                eb += bond_type[2 * t + 1] * dr * dr;
            }
        }
    }

    // ---- Angles: k * (theta - theta0)^2 ---- (indices streamed from global)
    {
        const int chunk = NANG_C / SPLIT; // 512
        const int i0 = s * chunk;
        for (int i = i0 + tid; i < i0 + chunk; i += THREADS) {
            const float* rp = feat + (size_t)(4 * i) * 9 + 7;
            if (i + THREADS < i0 + chunk)
                __builtin_prefetch(rp + (size_t)4 * 9 * THREADS, 0, 1); // global_prefetch_b8
            int ai = (int)rp[0];
            int aj = (int)rp[9];
            int ak = (int)rp[18];
            int t  = (int)rp[27];
            if (i < nang) {
                F3 qi = {sc[3 * ai], sc[3 * ai + 1], sc[3 * ai + 2]};
                F3 qj = {sc[3 * aj], sc[3 * aj + 1], sc[3 * aj + 2]};
                F3 qk = {sc[3 * ak], sc[3 * ak + 1], sc[3 * ak + 2]};
                F3 u = f3sub(qi, qj);
                F3 v = f3sub(qk, qj);
                float c = f3dot(u, v) * rsqrtf(f3dot(u, u) * f3dot(v, v));
                c = fminf(fmaxf(c, -0.9999f), 0.9999f);
                float theta = acosf(c);
                float dt = theta - angle_type[2 * t];
                ea += angle_type[2 * t + 1] * dt * dt;
            }
        }
    }

    // ---- Torsions: k * (1 + cos(n*phi - phi0)) ---- (indices from LDS via TDM)
    {
        for (int li = tid; li < TCHUNK; li += THREADS) {
            const int i = ti0 + li;
            const float* rp = &st[5 * li];
            int a0 = (int)rp[0];
            int a1 = (int)rp[1];
            int a2 = (int)rp[2];
            int a3 = (int)rp[3];
            int t  = (int)rp[4];
            if (i < nt) {
                F3 p0 = {sc[3 * a0], sc[3 * a0 + 1], sc[3 * a0 + 2]};
                F3 p1 = {sc[3 * a1], sc[3 * a1 + 1], sc[3 * a1 + 2]};
                F3 p2 = {sc[3 * a2], sc[3 * a2 + 1], sc[3 * a2 + 2]};
                F3 p3 = {sc[3 * a3], sc[3 * a3 + 1], sc[3 * a3 + 2]};
                F3 b1 = f3sub(p1, p0);
                F3 b2 = f3sub(p2, p1);
                F3 b3 = f3sub(p3, p2);
                F3 n1 = f3cross(b1, b2);
                F3 n2 = f3cross(b2, b3);
                F3 m1 = f3cross(n1, f3scale(b2, rsqrtf(f3dot(b2, b2))));
                float phi = atan2f(f3dot(m1, n2), f3dot(n1, n2));
                float k   = tor_type[3 * t];
                float nn  = tor_type[3 * t + 1];
                float ph0 = tor_type[3 * t + 2];
                et += k * (1.0f + cosf(nn * phi - ph0));
            }
        }
    }

    // ---- Deterministic block reduction ----
    for (int off = 16; off > 0; off >>= 1) {
        eb += __shfl_down(eb, off, 32);
        ea += __shfl_down(ea, off, 32);
        et += __shfl_down(et, off, 32);
    }
    if ((tid & 31) == 0) {
        red[tid >> 5][0] = eb;
        red[tid >> 5][1] = ea;
        red[tid >> 5][2] = et;
    }
    __syncthreads();
    if (tid == 0) {
        float sb = 0.0f, sa = 0.0f, stt = 0.0f;
        for (int w = 0; w < WAVES; ++w) { sb += red[w][0]; sa += red[w][1]; stt += red[w][2]; }
        float* o = ws + (size_t)((b * SPLIT + s) * 3);
        o[0] = sb; o[1] = sa; o[2] = stt;
    }
}

__global__ void finalize_kernel(const float* __restrict__ ws,
                                const float* __restrict__ opt,
                                float*       __restrict__ out)
{
    int idx = (int)threadIdx.x + (int)blockIdx.x * (int)blockDim.x;
    if (idx < B_C * 3) {
        int b = idx / 3, k = idx % 3;
        float sum = 0.0f;
        for (int q = 0; q < SPLIT; ++q) sum += ws[(b * SPLIT + q) * 3 + k];
        out[idx] = sum * opt[k];
    }
}

extern "C" void kernel_launch(void* const* d_in, const int* in_sizes, int n_in,
                              void* d_out, int out_size, void* d_ws, size_t ws_size,
                              hipStream_t stream)
{
    (void)in_sizes; (void)n_in; (void)out_size; (void)ws_size;
    const float* features   = (const float*)d_in[0];
    const int*   lengths    = (const int*)d_in[1];
    const float* bond_type  = (const float*)d_in[2];
    const float* angle_type = (const float*)d_in[3];
    const float* tor_type   = (const float*)d_in[4];
    const float* opt_pars   = (const float*)d_in[5];
    float* out = (float*)d_out;
    float* ws  = (float*)d_ws;

    energy_kernel<<<B_C * SPLIT, THREADS, 0, stream>>>(
        features, lengths, bond_type, angle_type, tor_type, ws);
    finalize_kernel<<<1, 256, 0, stream>>>(ws, opt_pars, out);
}